// mamba_23235773071546
// MI455X (gfx1250) — compile-verified
//
#include <hip/hip_runtime.h>
#include <hip/hip_bf16.h>

typedef __attribute__((ext_vector_type(2))) float v2f;
typedef __attribute__((ext_vector_type(8))) float v8f;

#define B_SZ    8
#define L_SEQ   1024
#define ENC_IN  32
#define D_MODEL 256
#define D_INNER 512
#define D_STATE 16
#define D_CONV  4
#define DT_RANK 16
#define D_FF    1024
#define BL      (B_SZ * L_SEQ)          // 8192 rows
#define XP_OUT  (DT_RANK + 2 * D_STATE) // 48

// ---------------------------------------------------------------------------
// fp32 WMMA GEMM with 1xNBLK register blocking:
//   C[M,N] = A[M,K] * W[N,K]^T  (+bias[n]) (+softplus)
// One wave computes a 16 x (16*NBLK) strip: one A fragment is reused across
// NBLK W fragments per K-step (V_WMMA_F32_16X16X4_F32, K step 4).
// A-frag: lane holds A[m][kb..kb+1], m = lane&15, kb = k + 2*(lane>>4)
// B-frag: lane holds W[n][kb..kb+1], n = lane&15  (W row-major (N,K) == B^T)
// D-frag: acc[j][v] is C[tM*16 + v + 8*(lane>>4)][nbase + j*16 + (lane&15)]
// ---------------------------------------------------------------------------
template <int NBLK>
__global__ __launch_bounds__(256) void wmma_gemm_nt(
    const float* __restrict__ A, int lda,
    const float* __restrict__ W,           // (N, K) row-major weights
    const float* __restrict__ bias,        // may be nullptr
    float* __restrict__ C, int ldc,
    int M, int N, int K, int softplus_act)
{
    const int lane    = threadIdx.x & 31;
    const int wave    = threadIdx.x >> 5;
    const int tilesNB = (N >> 4) / NBLK;
    const int tile    = blockIdx.x * 8 + wave;
    const int tNB     = tile % tilesNB;
    const int tM      = tile / tilesNB;
    if (tM >= (M >> 4)) return;            // wave-uniform guard

    const int half = lane >> 4;            // 0 or 1 -> K sub-pair
    const int lidx = lane & 15;

    const float* Arow = A + (size_t)(tM * 16 + lidx) * lda;
    const float* Wrow = W + (size_t)(tNB * NBLK * 16 + lidx) * K;

    v8f acc[NBLK] = {};
#pragma unroll 2
    for (int k = 0; k < K; k += 4) {
        const int kb = k + 2 * half;
        const v2f av = *(const v2f*)(Arow + kb);
#pragma unroll
        for (int j = 0; j < NBLK; ++j) {
            const v2f bv = *(const v2f*)(Wrow + (size_t)j * 16 * K + kb);
            acc[j] = __builtin_amdgcn_wmma_f32_16x16x4_f32(
                /*neg_a=*/false, av, /*neg_b=*/false, bv,
                /*c_mod=*/(short)0, acc[j], /*reuse_a=*/false, /*reuse_b=*/false);
        }
    }

#pragma unroll
    for (int j = 0; j < NBLK; ++j) {
        const int n  = tNB * NBLK * 16 + j * 16 + lidx;
        const float bn = bias ? bias[n] : 0.0f;
#pragma unroll
        for (int v = 0; v < 8; ++v) {
            const int m = tM * 16 + v + 8 * half;
            float x = acc[j][v] + bn;
            if (softplus_act) x = (x > 20.0f) ? x : log1pf(__expf(x));
            C[(size_t)m * ldc + n] = x;
        }
    }
}

// ---------------------------------------------------------------------------
// Causal depthwise conv (D_CONV=4) + bias + SiLU on xi = xz[..., :512]
// ---------------------------------------------------------------------------
__global__ __launch_bounds__(256) void conv_silu_kernel(
    const float* __restrict__ xz, const float* __restrict__ cw,
    const float* __restrict__ cb, float* __restrict__ xc)
{
    const int idx = blockIdx.x * blockDim.x + threadIdx.x;   // B*L*D_INNER
    const int d = idx & (D_INNER - 1);
    const int t = (idx >> 9) & (L_SEQ - 1);
    const int b = idx >> 19;
    const float* xi = xz + (size_t)b * L_SEQ * (2 * D_INNER);
    float acc = cb[d];
#pragma unroll
    for (int j = 0; j < D_CONV; ++j) {
        const int tt = t - (D_CONV - 1) + j;
        if (tt >= 0)
            acc += cw[d * D_CONV + j] * xi[(size_t)tt * (2 * D_INNER) + d];
    }
    acc = acc / (1.0f + __expf(-acc));                        // SiLU
    xc[idx] = acc;
}

// ---------------------------------------------------------------------------
// Selective scan, fused with D-skip and SiLU(z) gating.
// Thread = (b, d, state-quarter q): 4 states in registers per lane,
// 16384 threads = 512 waves. Per-step y reduced across the 4 quarter-lanes
// with two xor-shuffles; no LDS, no barriers. B/C quarters are contiguous
// 16B-aligned float4 loads from xdbl (L1 broadcasts shared reads).
// ---------------------------------------------------------------------------
__global__ __launch_bounds__(256) void scan_kernel(
    const float* __restrict__ xc,     // u           (BL, 512)
    const float* __restrict__ dtb,    // softplus dt (BL, 512)
    const float* __restrict__ xdbl,   // (BL, 48): [0:16)=dt_lo [16:32)=B [32:48)=C
    const float* __restrict__ xz,     // z at [..., 512:1024)
    const float* __restrict__ A_log,  // (512, 16) for this layer
    const float* __restrict__ Dsk,    // (512,)
    float* __restrict__ y)            // (BL, 512)
{
    const int tid  = threadIdx.x;
    const int q    = tid & 3;                 // state quarter 0..3
    const int dloc = tid >> 2;                // 0..63
    const int d    = blockIdx.x * 64 + dloc;
    const int b    = blockIdx.y;

    const float4 al = *(const float4*)(A_log + d * D_STATE + q * 4);
    const float An0 = -__expf(al.x), An1 = -__expf(al.y);
    const float An2 = -__expf(al.z), An3 = -__expf(al.w);
    const float Dv = Dsk[d];

    float h0 = 0.f, h1 = 0.f, h2 = 0.f, h3 = 0.f;
    const float* xdr = xdbl + DT_RANK + q * 4;   // + row*48 below

    for (int t = 0; t < L_SEQ; ++t) {
        const size_t row = (size_t)b * L_SEQ + t;
        const float dtv = dtb[row * D_INNER + d];
        const float u   = xc [row * D_INNER + d];
        const float4 Bq = *(const float4*)(xdr + row * XP_OUT);
        const float4 Cq = *(const float4*)(xdr + row * XP_OUT + D_STATE);
        const float du = dtv * u;
        h0 = __expf(dtv * An0) * h0 + du * Bq.x;
        h1 = __expf(dtv * An1) * h1 + du * Bq.y;
        h2 = __expf(dtv * An2) * h2 + du * Bq.z;
        h3 = __expf(dtv * An3) * h3 + du * Bq.w;
        float acc = h0 * Cq.x + h1 * Cq.y + h2 * Cq.z + h3 * Cq.w;
        acc += __shfl_xor(acc, 1, 32);
        acc += __shfl_xor(acc, 2, 32);
        if (q == 0) {
            acc += u * Dv;
            const float zv = xz[row * (2 * D_INNER) + D_INNER + d];
            acc *= zv / (1.0f + __expf(-zv));                 // * SiLU(z)
            y[row * D_INNER + d] = acc;
        }
    }
}

// ---------------------------------------------------------------------------
// Head: p1 (exact GELU) then p2. M=8 — too small for 16-row WMMA tiles.
// ---------------------------------------------------------------------------
__global__ __launch_bounds__(256) void head_p1_kernel(
    const float* __restrict__ hbuf, const float* __restrict__ p1w,
    const float* __restrict__ p1b, float* __restrict__ g)
{
    const int idx = blockIdx.x * blockDim.x + threadIdx.x;   // B_SZ * D_FF
    const int f = idx & (D_FF - 1);
    const int b = idx >> 10;
    const float* hr = hbuf + ((size_t)b * L_SEQ + (L_SEQ - 1)) * D_MODEL;
    const float* wr = p1w + (size_t)f * D_MODEL;
    float acc = p1b[f];
    for (int k = 0; k < D_MODEL; ++k) acc += hr[k] * wr[k];
    acc = 0.5f * acc * (1.0f + erff(acc * 0.70710678118654752f));
    g[idx] = acc;
}

__global__ __launch_bounds__(256) void head_p2_kernel(
    const float* __restrict__ g, const float* __restrict__ p2w,
    const float* __restrict__ p2b, float* __restrict__ out)
{
    __shared__ float red[256];
    const int b = blockIdx.x;
    float acc = 0.0f;
    for (int f = threadIdx.x; f < D_FF; f += 256) acc += g[b * D_FF + f] * p2w[f];
    red[threadIdx.x] = acc;
    __syncthreads();
    for (int s = 128; s > 0; s >>= 1) {
        if (threadIdx.x < s) red[threadIdx.x] += red[threadIdx.x + s];
        __syncthreads();
    }
    if (threadIdx.x == 0) out[b] = red[0] + p2b[0];
}

// ---------------------------------------------------------------------------

static inline int gemm_grid(int M, int N, int nblk) {
    return (M / 16) * ((N / 16) / nblk) / 8;
}

extern "C" void kernel_launch(void* const* d_in, const int* in_sizes, int n_in,
                              void* d_out, int out_size, void* d_ws, size_t ws_size,
                              hipStream_t stream) {
    const float* x     = (const float*)d_in[0];
    const float* in_w  = (const float*)d_in[1];
    const float* in_b  = (const float*)d_in[2];
    const float* ipw   = (const float*)d_in[3];   // (2, 1024, 256)
    const float* cw    = (const float*)d_in[4];   // (2, 512, 4)
    const float* cb    = (const float*)d_in[5];   // (2, 512)
    const float* xpw   = (const float*)d_in[6];   // (2, 48, 512)
    const float* dtw   = (const float*)d_in[7];   // (2, 512, 16)
    const float* dt_b  = (const float*)d_in[8];   // (2, 512)
    const float* A_log = (const float*)d_in[9];   // (2, 512, 16)
    const float* Dsk   = (const float*)d_in[10];  // (2, 512)
    const float* opw   = (const float*)d_in[11];  // (2, 256, 512)
    const float* p1w   = (const float*)d_in[12];
    const float* p1b   = (const float*)d_in[13];
    const float* p2w   = (const float*)d_in[14];
    const float* p2b   = (const float*)d_in[15];
    float* out = (float*)d_out;

    // Workspace layout (floats)
    float* ws = (float*)d_ws;
    size_t off = 0;
    float* hbuf  = ws + off; off += (size_t)BL * D_MODEL;       // 8192*256
    float* xzb   = ws + off; off += (size_t)BL * 2 * D_INNER;   // 8192*1024
    float* xcb   = ws + off; off += (size_t)BL * D_INNER;       // 8192*512
    float* xdblb = ws + off; off += (size_t)BL * XP_OUT;        // 8192*48
    float* dtbuf = ws + off; off += (size_t)BL * D_INNER;       // 8192*512
    float* ybuf  = ws + off; off += (size_t)BL * D_INNER;       // 8192*512
    float* gbuf  = ws + off; off += (size_t)B_SZ * D_FF;        // 8*1024

    // Embed: h = x @ in_w^T + in_b   (8192 x 256, K=32)
    wmma_gemm_nt<4><<<gemm_grid(BL, D_MODEL, 4), 256, 0, stream>>>(
        x, ENC_IN, in_w, in_b, hbuf, D_MODEL, BL, D_MODEL, ENC_IN, 0);

    for (int layer = 0; layer < 2; ++layer) {
        const float* ipw_l = ipw   + (size_t)layer * 2 * D_INNER * D_MODEL;
        const float* cw_l  = cw    + (size_t)layer * D_INNER * D_CONV;
        const float* cb_l  = cb    + (size_t)layer * D_INNER;
        const float* xpw_l = xpw   + (size_t)layer * XP_OUT * D_INNER;
        const float* dtw_l = dtw   + (size_t)layer * D_INNER * DT_RANK;
        const float* dtb_l = dt_b  + (size_t)layer * D_INNER;
        const float* Al_l  = A_log + (size_t)layer * D_INNER * D_STATE;
        const float* Dsk_l = Dsk   + (size_t)layer * D_INNER;
        const float* opw_l = opw   + (size_t)layer * D_MODEL * D_INNER;

        // in_proj: xz = h @ ipw^T   (8192 x 1024, K=256)
        wmma_gemm_nt<4><<<gemm_grid(BL, 2 * D_INNER, 4), 256, 0, stream>>>(
            hbuf, D_MODEL, ipw_l, nullptr, xzb, 2 * D_INNER,
            BL, 2 * D_INNER, D_MODEL, 0);

        // causal depthwise conv + bias + SiLU -> xc
        conv_silu_kernel<<<(BL * D_INNER) / 256, 256, 0, stream>>>(
            xzb, cw_l, cb_l, xcb);

        // x_proj: xdbl = xc @ xpw^T   (8192 x 48, K=512)
        wmma_gemm_nt<1><<<gemm_grid(BL, XP_OUT, 1), 256, 0, stream>>>(
            xcb, D_INNER, xpw_l, nullptr, xdblb, XP_OUT,
            BL, XP_OUT, D_INNER, 0);

        // dt = softplus(dt_lo @ dtw^T + dtb)   (8192 x 512, K=16; A lda=48)
        wmma_gemm_nt<4><<<gemm_grid(BL, D_INNER, 4), 256, 0, stream>>>(
            xdblb, XP_OUT, dtw_l, dtb_l, dtbuf, D_INNER,
            BL, D_INNER, DT_RANK, 1);

        // selective scan fused with D-skip + SiLU(z) gating -> y
        scan_kernel<<<dim3(D_INNER / 64, B_SZ), 256, 0, stream>>>(
            xcb, dtbuf, xdblb, xzb, Al_l, Dsk_l, ybuf);

        // out_proj: h = y @ opw^T   (8192 x 256, K=512)
        wmma_gemm_nt<4><<<gemm_grid(BL, D_MODEL, 4), 256, 0, stream>>>(
            ybuf, D_INNER, opw_l, nullptr, hbuf, D_MODEL,
            BL, D_MODEL, D_INNER, 0);
    }

    // Head: last timestep -> p1 (gelu) -> p2
    head_p1_kernel<<<(B_SZ * D_FF) / 256, 256, 0, stream>>>(hbuf, p1w, p1b, gbuf);
    head_p2_kernel<<<B_SZ, 256, 0, stream>>>(gbuf, p2w, p2b, out);
}